// GNNLayer_41686952575549
// MI455X (gfx1250) — compile-verified
//
#include <hip/hip_runtime.h>
#include <hip/hip_bf16.h>

#define D_FEAT 128
#define NEG_SLOPE 0.01f
#define LDS_PITCH 130   // 128 + 2 pad: row stride of 130 floats -> conflict-free ds_load_b64

typedef float v2f __attribute__((ext_vector_type(2)));
typedef float v8f __attribute__((ext_vector_type(8)));

// ---------------- kernel 1: zero h_neigh workspace ----------------
__global__ __launch_bounds__(256) void zero_ws_kernel(float4* __restrict__ ws, int n4) {
    int i = blockIdx.x * blockDim.x + threadIdx.x;
    if (i < n4) ws[i] = make_float4(0.f, 0.f, 0.f, 0.f);
}

// ---------------- kernel 2: gather-scale-scatter over edges ----------------
// 32 lanes per edge; each lane handles a float4 chunk of the 128-wide row.
__global__ __launch_bounds__(256) void edge_scatter_kernel(
    const float* __restrict__ feat,
    const int*   __restrict__ target,
    const int*   __restrict__ neighbor,
    const float* __restrict__ values,
    float*       __restrict__ hneigh,
    int E)
{
    int t = blockIdx.x * blockDim.x + threadIdx.x;
    int e = t >> 5;          // edge id
    int c = t & 31;          // float4 chunk within the row
    if (e >= E) return;

    int   nbr = neighbor[e];
    int   tgt = target[e];
    float v   = values[e];

    const float4* src = (const float4*)(feat + (size_t)nbr * D_FEAT);
    float4 f = src[c];

    float* dst = hneigh + (size_t)tgt * D_FEAT + (c << 2);
    unsafeAtomicAdd(dst + 0, f.x * v);   // -> global_atomic_add_f32 (L2-resident)
    unsafeAtomicAdd(dst + 1, f.y * v);
    unsafeAtomicAdd(dst + 2, f.z * v);
    unsafeAtomicAdd(dst + 3, f.w * v);
}

// ---------------- kernel 3: fused dual-GEMM + bias + leaky relu ----------------
// One block = 16 output rows (nodes). 8 waves, each owns one 16-wide column tile.
// out[16x128] = X1[16x128] @ W1^T + X2[16x128] @ W2^T + (b1+b2), leaky relu.
__global__ __launch_bounds__(256) void gemm_fused_kernel(
    const float* __restrict__ feat,
    const float* __restrict__ hneigh,
    const float* __restrict__ W1,
    const float* __restrict__ b1,
    const float* __restrict__ W2,
    const float* __restrict__ b2,
    float*       __restrict__ out,
    int nrows)
{
    __shared__ float X1[16 * LDS_PITCH];
    __shared__ float X2[16 * LDS_PITCH];

    const int rowbase = blockIdx.x * 16;
    const int tid = threadIdx.x;

    // ---- stage X1 = f + hn, X2 = f * hn into LDS (coalesced along columns) ----
    for (int idx = tid; idx < 16 * D_FEAT; idx += 256) {
        int r = idx >> 7;          // 0..15
        int c = idx & 127;         // 0..127
        int gr = rowbase + r;
        float f = 0.f, hn = 0.f;
        if (gr < nrows) {
            f  = feat  [(size_t)gr * D_FEAT + c];
            hn = hneigh[(size_t)gr * D_FEAT + c];
        }
        X1[r * LDS_PITCH + c] = f + hn;
        X2[r * LDS_PITCH + c] = f * hn;
    }
    __syncthreads();

    // ---- per-wave 16x16 output tile via V_WMMA_F32_16X16X4_F32 ----
    const int wave = tid >> 5;
    const int lane = tid & 31;
    const int half = lane >> 4;    // 0: K={0,1}; 1: K={2,3}
    const int lr   = lane & 15;    // A: row M; B: col N
    const int col  = wave * 16 + lr;

    const float bias = b1[col] + b2[col];
    v8f acc = { bias, bias, bias, bias, bias, bias, bias, bias };

    // A fragment source: row lr of LDS tile, K offset 2*half
    const float* x1p = &X1[lr * LDS_PITCH + 2 * half];
    const float* x2p = &X2[lr * LDS_PITCH + 2 * half];
    // B fragment source: B[k][n] = W[n][k] -> row `col` of W, K offset 2*half (8B aligned)
    const float* w1p = W1 + (size_t)col * D_FEAT + 2 * half;
    const float* w2p = W2 + (size_t)col * D_FEAT + 2 * half;

    #pragma unroll
    for (int k = 0; k < D_FEAT; k += 4) {
        v2f a1 = { x1p[k], x1p[k + 1] };
        v2f a2 = { x2p[k], x2p[k + 1] };
        v2f bw1 = { w1p[k], w1p[k + 1] };
        v2f bw2 = { w2p[k], w2p[k + 1] };
        acc = __builtin_amdgcn_wmma_f32_16x16x4_f32(
                  false, a1, false, bw1, (short)0, acc, false, false);
        acc = __builtin_amdgcn_wmma_f32_16x16x4_f32(
                  false, a2, false, bw2, (short)0, acc, false, false);
    }

    // ---- epilogue: leaky relu + store (C/D layout: VGPR v -> M = v + 8*half) ----
    #pragma unroll
    for (int v = 0; v < 8; v++) {
        int row = rowbase + v + 8 * half;
        if (row < nrows) {
            float h = acc[v];
            out[(size_t)row * D_FEAT + col] = (h >= 0.f) ? h : NEG_SLOPE * h;
        }
    }
}

// ---------------- host-side launcher ----------------
extern "C" void kernel_launch(void* const* d_in, const int* in_sizes, int n_in,
                              void* d_out, int out_size, void* d_ws, size_t ws_size,
                              hipStream_t stream) {
    const float* feat     = (const float*)d_in[0];   // (N, 128)
    const int*   target   = (const int*)  d_in[1];   // (E,)
    const int*   neighbor = (const int*)  d_in[2];   // (E,)
    const float* values   = (const float*)d_in[3];   // (E,)
    const float* W1       = (const float*)d_in[4];   // (128, 128)
    const float* b1       = (const float*)d_in[5];   // (128,)
    const float* W2       = (const float*)d_in[6];   // (128, 128)
    const float* b2       = (const float*)d_in[7];   // (128,)

    const int N = in_sizes[0] / D_FEAT;              // 50000
    const int E = in_sizes[1];                       // 625000

    float* hneigh = (float*)d_ws;                    // N*128 floats (25.6 MB)
    float* outp   = (float*)d_out;

    // 1) zero the scatter accumulator (must happen every call)
    {
        int n4 = (N * D_FEAT) / 4;
        int blocks = (n4 + 255) / 256;
        zero_ws_kernel<<<blocks, 256, 0, stream>>>((float4*)hneigh, n4);
    }
    // 2) edge gather-scale-scatter (fp32 HW atomics, L2-resident target)
    {
        long long threads = (long long)E * 32;
        int blocks = (int)((threads + 255) / 256);
        edge_scatter_kernel<<<blocks, 256, 0, stream>>>(feat, target, neighbor,
                                                        values, hneigh, E);
    }
    // 3) fused dual-GEMM + bias + leaky relu via fp32 WMMA
    {
        int blocks = (N + 15) / 16;                  // 3125
        gemm_fused_kernel<<<blocks, 256, 0, stream>>>(feat, hneigh, W1, b1, W2, b2,
                                                      outp, N);
    }
}